// AttentionManifold_7791070675909
// MI455X (gfx1250) — compile-verified
//
#include <hip/hip_runtime.h>
#include <hip/hip_bf16.h>

typedef __attribute__((ext_vector_type(2))) float v2f;
typedef __attribute__((ext_vector_type(8))) float v8f;

// D = A(16x4,f32) * B(4x16,f32) + C, wave32 CDNA5 WMMA
static __device__ __forceinline__ v8f wmma_f32_k4(v2f a, v2f b, v8f c) {
  return __builtin_amdgcn_wmma_f32_16x16x4_f32(false, a, false, b, (short)0, c,
                                               false, false);
}

// ---------------------------------------------------------------------------
// Kernel 1: S = W^T X W  for each (b,m) and W in {Wq,Wk,Wv}
//   grid = 3*2048 blocks, 256 threads (8 waves). X streamed from global
//   (X is 82MB < 192MB L2, so the 4x reread per N-tile stays on-chip),
//   W and T staged in LDS. Padding rows handled by clamp+mask (no EXEC
//   divergence around the loads -> unconditional global_load_b64).
// ---------------------------------------------------------------------------
__global__ __launch_bounds__(256) void congruence_kernel(
    const float* __restrict__ x,                       // [2048,100,100]
    const float* __restrict__ Wq, const float* __restrict__ Wk,
    const float* __restrict__ Wv,
    float* __restrict__ Sout)                          // [3,2048,64,64]
{
  __shared__ float lw[100 * 64];     // W, row-major [k][n]
  __shared__ float lt[112 * 64];     // T = X*W, rows padded to 112

  const int bid = blockIdx.x;
  const int w   = bid >> 11;         // which weight
  const int bm  = bid & 2047;
  const float* W  = (w == 0) ? Wq : ((w == 1) ? Wk : Wv);
  const float* xg = x + (size_t)bm * 10000;
  const int tid = threadIdx.x;

  for (int i = tid; i < 6400; i += 256) lw[i] = W[i];
  __syncthreads();

  const int wave = tid >> 5, lane = tid & 31;
  const int kh = lane >> 4, l16 = lane & 15;   // lane-half, lane-in-half

  // GEMM1: T[112x64] = Xpad[112x100] * W[100x64]; 7x4 = 28 tiles
  for (int tile = wave; tile < 28; tile += 8) {
    const int tm = (tile >> 2) * 16, tn = (tile & 3) * 16;
    const int row = tm + l16;
    const int rowc = row < 100 ? row : 99;     // clamp: load stays in-bounds
    const float msk = row < 100 ? 1.0f : 0.0f; // zero contribution of pad rows
    const float* xr = xg + rowc * 100;
    v8f acc = {};
    for (int k = 0; k < 100; k += 4) {
      const int ka = k + 2 * kh;
      v2f a, b2;
      a.x  = xr[ka] * msk;                     // contiguous pair -> b64 load
      a.y  = xr[ka + 1] * msk;
      b2.x = lw[ka * 64 + tn + l16];
      b2.y = lw[(ka + 1) * 64 + tn + l16];
      acc = wmma_f32_k4(a, b2, acc);
    }
    for (int r = 0; r < 8; r++)
      lt[(tm + r + 8 * kh) * 64 + tn + l16] = acc[r];
  }
  __syncthreads();

  // GEMM2: S[64x64] = W^T[64x100] * T[100x64]; 4x4 = 16 tiles
  float* Sg = Sout + ((size_t)w * 2048 + bm) * 4096;
  for (int tile = wave; tile < 16; tile += 8) {
    const int tm = (tile >> 2) * 16, tn = (tile & 3) * 16;
    v8f acc = {};
    for (int k = 0; k < 100; k += 4) {
      const int ka = k + 2 * kh;
      v2f a, b2;
      a.x  = lw[ka * 64 + tm + l16];           // A[m][k] = W[k][m]
      a.y  = lw[(ka + 1) * 64 + tm + l16];
      b2.x = lt[ka * 64 + tn + l16];
      b2.y = lt[(ka + 1) * 64 + tn + l16];
      acc = wmma_f32_k4(a, b2, acc);
    }
    for (int r = 0; r < 8; r++)
      Sg[(tm + r + 8 * kh) * 64 + tn + l16] = acc[r];
  }
}

// ---------------------------------------------------------------------------
// Kernel 2: in-place  M <- U f(Lambda) U^T  via one-sided (Hestenes) Jacobi.
//   mode 0: f = log(|l|+1e-6)   mode 1: f = exp(l)
//   One workgroup (256 thr) per matrix. G = A*V invariant; lambda_j = v_j.g_j.
//   Reconstruction (V*D)*V^T uses fp32 WMMA. norms[id] = ||f(M)||_F^2.
// ---------------------------------------------------------------------------
__global__ __launch_bounds__(256) void jacobi_fn_kernel(
    float* __restrict__ mats,        // [N,64,64] in/out
    float* __restrict__ norms,       // [N] or nullptr
    const int mode)
{
  __shared__ float G[64 * 65];       // stride 65 -> conflict-free columns
  __shared__ float V[64 * 65];
  __shared__ float red[8];

  const int id = blockIdx.x;
  float* M = mats + (size_t)id * 4096;
  const int tid = threadIdx.x;

  for (int i = tid; i < 4096; i += 256) {
    const int r = i >> 6, c = i & 63;
    G[r * 65 + c] = M[i];
    V[r * 65 + c] = (r == c) ? 1.0f : 0.0f;
  }
  __syncthreads();

  const int pair = tid >> 3, t8 = tid & 7;   // 32 pairs x 8 threads (same wave)
  for (int sweep = 0; sweep < 8; sweep++) {
    for (int rnd = 0; rnd < 63; rnd++) {
      int p, q;
      if (pair == 0) { p = 63; q = rnd; }
      else {
        p = (rnd + pair) % 63;
        q = (rnd + 63 - pair) % 63;
      }
      // column dot products (8-lane split, shuffle reduce within wave)
      float aa = 0.f, bb = 0.f, ab = 0.f;
      for (int rr = t8; rr < 64; rr += 8) {
        const float gp = G[rr * 65 + p], gq = G[rr * 65 + q];
        aa += gp * gp; bb += gq * gq; ab += gp * gq;
      }
      for (int o = 4; o > 0; o >>= 1) {
        aa += __shfl_xor(aa, o, 8);
        bb += __shfl_xor(bb, o, 8);
        ab += __shfl_xor(ab, o, 8);
      }
      float c = 1.0f, s = 0.0f;
      if (fabsf(ab) > 1e-30f) {
        const float zeta = (bb - aa) / (2.0f * ab);
        const float t = copysignf(1.0f, zeta) /
                        (fabsf(zeta) + sqrtf(1.0f + zeta * zeta));
        c = rsqrtf(1.0f + t * t);
        s = c * t;
      }
      for (int rr = t8; rr < 64; rr += 8) {
        const float gp = G[rr * 65 + p], gq = G[rr * 65 + q];
        G[rr * 65 + p] = c * gp - s * gq;
        G[rr * 65 + q] = s * gp + c * gq;
        const float vp = V[rr * 65 + p], vq = V[rr * 65 + q];
        V[rr * 65 + p] = c * vp - s * vq;
        V[rr * 65 + q] = s * vp + c * vq;
      }
      __syncthreads();
    }
  }

  // eigenvalues + scale columns of G <- V * f(lambda)
  {
    const int col = tid >> 2, t4 = tid & 3;  // 64 cols x 4 threads
    float lam = 0.f;
    for (int rr = t4; rr < 64; rr += 4)
      lam += V[rr * 65 + col] * G[rr * 65 + col];
    for (int o = 2; o > 0; o >>= 1) lam += __shfl_xor(lam, o, 4);
    const float fl = (mode == 0) ? logf(fabsf(lam) + 1e-6f) : expf(lam);
    for (int rr = t4; rr < 64; rr += 4)
      G[rr * 65 + col] = V[rr * 65 + col] * fl;
  }
  __syncthreads();

  // M = (V*D) * V^T via WMMA; accumulate Frobenius norm of result
  const int wave = tid >> 5, lane = tid & 31;
  const int kh = lane >> 4, l16 = lane & 15;
  float nrm = 0.f;
  for (int tile = wave; tile < 16; tile += 8) {
    const int tm = (tile >> 2) * 16, tn = (tile & 3) * 16;
    v8f acc = {};
    for (int k = 0; k < 64; k += 4) {
      const int ka = k + 2 * kh;
      v2f a, b2;
      a.x  = G[(tm + l16) * 65 + ka];
      a.y  = G[(tm + l16) * 65 + ka + 1];
      b2.x = V[(tn + l16) * 65 + ka];      // B[k][n] = V[n][k]
      b2.y = V[(tn + l16) * 65 + ka + 1];
      acc = wmma_f32_k4(a, b2, acc);
    }
    for (int r = 0; r < 8; r++) {
      const float v = acc[r];
      M[(tm + r + 8 * kh) * 64 + tn + l16] = v;
      nrm += v * v;
    }
  }
  if (norms) {
    for (int o = 16; o > 0; o >>= 1) nrm += __shfl_xor(nrm, o, 32);
    if (lane == 0) red[wave] = nrm;
    __syncthreads();
    if (tid == 0) {
      float t = 0.f;
      for (int i = 0; i < 8; i++) t += red[i];
      norms[id] = t;
    }
  }
}

// ---------------------------------------------------------------------------
// Kernel 3: per-b attention. cross = kf . qf^T (64x64x4096 WMMA GEMM via LDS
// K-chunks), then scores -> softmax over K-index -> transposed prob[b][j][i].
// ---------------------------------------------------------------------------
__global__ __launch_bounds__(256) void attention_kernel(
    const float* __restrict__ logK,   // [2048,4096]
    const float* __restrict__ logQ,   // [2048,4096]
    const float* __restrict__ norms,  // [0..2048)=qn, [2048..4096)=kn
    float* __restrict__ prob)         // [32,64,64]: prob[b][j][i]
{
  __shared__ float lk[64 * 33];
  __shared__ float lq[64 * 33];
  __shared__ float sc[64 * 65];
  __shared__ float qn_s[64], kn_s[64];

  const int b = blockIdx.x;
  const int tid = threadIdx.x;
  if (tid < 64) {
    qn_s[tid] = norms[b * 64 + tid];
    kn_s[tid] = norms[2048 + b * 64 + tid];
  }
  const int wave = tid >> 5, lane = tid & 31;
  const int kh = lane >> 4, l16 = lane & 15;

  // each wave owns 2 of the 16 output tiles
  const int t0 = wave, t1 = wave + 8;
  const int tm0 = (t0 >> 2) * 16, tn0 = (t0 & 3) * 16;
  const int tm1 = (t1 >> 2) * 16, tn1 = (t1 & 3) * 16;
  v8f acc0 = {}, acc1 = {};

  const float* Kb = logK + (size_t)b * 64 * 4096;
  const float* Qb = logQ + (size_t)b * 64 * 4096;
  for (int k0 = 0; k0 < 4096; k0 += 32) {
    __syncthreads();
    for (int i = tid; i < 64 * 32; i += 256) {
      const int rr = i >> 5, cc = i & 31;
      lk[rr * 33 + cc] = Kb[(size_t)rr * 4096 + k0 + cc];
      lq[rr * 33 + cc] = Qb[(size_t)rr * 4096 + k0 + cc];
    }
    __syncthreads();
    for (int k = 0; k < 32; k += 4) {
      const int ka = k + 2 * kh;
      v2f a, b2;
      a.x  = lk[(tm0 + l16) * 33 + ka]; a.y  = lk[(tm0 + l16) * 33 + ka + 1];
      b2.x = lq[(tn0 + l16) * 33 + ka]; b2.y = lq[(tn0 + l16) * 33 + ka + 1];
      acc0 = wmma_f32_k4(a, b2, acc0);
      a.x  = lk[(tm1 + l16) * 33 + ka]; a.y  = lk[(tm1 + l16) * 33 + ka + 1];
      b2.x = lq[(tn1 + l16) * 33 + ka]; b2.y = lq[(tn1 + l16) * 33 + ka + 1];
      acc1 = wmma_f32_k4(a, b2, acc1);
    }
  }
  __syncthreads();

  // scores = 1/(1+log1p(E)), E = max(kn_i+qn_j-2*cross,0) + 64e-6
  for (int r = 0; r < 8; r++) {
    {
      const int i = tm0 + r + 8 * kh, j = tn0 + l16;
      const float E = fmaxf(kn_s[i] + qn_s[j] - 2.0f * acc0[r], 0.0f) + 64e-6f;
      sc[i * 65 + j] = 1.0f / (1.0f + log1pf(E));
    }
    {
      const int i = tm1 + r + 8 * kh, j = tn1 + l16;
      const float E = fmaxf(kn_s[i] + qn_s[j] - 2.0f * acc1[r], 0.0f) + 64e-6f;
      sc[i * 65 + j] = 1.0f / (1.0f + log1pf(E));
    }
  }
  __syncthreads();

  // softmax over i for each column j; write prob[b][j][i]
  const int col = tid >> 2, t4 = tid & 3;
  float mx = -1e30f;
  for (int i = t4; i < 64; i += 4) mx = fmaxf(mx, sc[i * 65 + col]);
  for (int o = 2; o > 0; o >>= 1) mx = fmaxf(mx, __shfl_xor(mx, o, 4));
  float sum = 0.f;
  for (int i = t4; i < 64; i += 4) sum += __expf(sc[i * 65 + col] - mx);
  for (int o = 2; o > 0; o >>= 1) sum += __shfl_xor(sum, o, 4);
  const float inv = 1.0f / sum;
  float* pb = prob + (size_t)b * 4096 + col * 64;
  for (int i = t4; i < 64; i += 4)
    pb[i] = __expf(sc[i * 65 + col] - mx) * inv;
}

// ---------------------------------------------------------------------------
// Kernel 4: mixed[b,j,:] = sum_i prob[b,j,i] * logV[b,i,:]  (WMMA, N-chunked)
// ---------------------------------------------------------------------------
__global__ __launch_bounds__(256) void mix_kernel(
    const float* __restrict__ prob,   // [32,64,64]
    const float* __restrict__ logV,   // [2048,4096]
    float* __restrict__ mixed)        // [2048,4096] (= d_out)
{
  __shared__ float lp[64 * 65];
  const int b  = blockIdx.x >> 4;
  const int n0 = (blockIdx.x & 15) * 256;
  const int tid = threadIdx.x;

  for (int i = tid; i < 4096; i += 256) {
    const int r = i >> 6, c = i & 63;
    lp[r * 65 + c] = prob[(size_t)b * 4096 + i];
  }
  __syncthreads();

  const int wave = tid >> 5, lane = tid & 31;
  const int kh = lane >> 4, l16 = lane & 15;
  const float* Vb = logV + (size_t)b * 64 * 4096;
  float* Ob = mixed + (size_t)b * 64 * 4096;

  // 4 (M) x 16 (N) tiles per chunk = 64 tiles / 8 waves
  for (int tile = wave; tile < 64; tile += 8) {
    const int tm = (tile >> 4) * 16;
    const int tn = n0 + (tile & 15) * 16;
    v8f acc = {};
    for (int k = 0; k < 64; k += 4) {
      const int ka = k + 2 * kh;
      v2f a, b2;
      a.x  = lp[(tm + l16) * 65 + ka];
      a.y  = lp[(tm + l16) * 65 + ka + 1];
      b2.x = Vb[(size_t)ka * 4096 + tn + l16];
      b2.y = Vb[(size_t)(ka + 1) * 4096 + tn + l16];
      acc = wmma_f32_k4(a, b2, acc);
    }
    for (int r = 0; r < 8; r++)
      Ob[(size_t)(tm + r + 8 * kh) * 4096 + tn + l16] = acc[r];
  }
}

// ---------------------------------------------------------------------------
extern "C" void kernel_launch(void* const* d_in, const int* in_sizes, int n_in,
                              void* d_out, int out_size, void* d_ws,
                              size_t ws_size, hipStream_t stream) {
  const float* x  = (const float*)d_in[0];
  const float* Wq = (const float*)d_in[1];
  const float* Wk = (const float*)d_in[2];
  const float* Wv = (const float*)d_in[3];
  float* out = (float*)d_out;

  const size_t mats = (size_t)3 * 2048 * 4096;       // 100.7 MB
  float* S     = (float*)d_ws;                       // [3][2048][64*64]
  float* norms = S + mats;                           // [6144]
  float* prob  = norms + 6144;                       // [32*64*64]

  // 1) Q,K,V = W^T X W           (fp32 WMMA)
  congruence_kernel<<<6144, 256, 0, stream>>>(x, Wq, Wk, Wv, S);
  // 2) logQ/logK/logV in place   (Jacobi + WMMA reconstruction) + norms
  jacobi_fn_kernel<<<6144, 256, 0, stream>>>(S, norms, 0);
  // 3) scores + softmax -> prob  (fp32 WMMA cross GEMM)
  attention_kernel<<<32, 256, 0, stream>>>(
      S + (size_t)1 * 2048 * 4096,   // logK
      S,                             // logQ
      norms, prob);
  // 4) mixed = prob . logV -> d_out (fp32 WMMA)
  mix_kernel<<<512, 256, 0, stream>>>(prob, S + (size_t)2 * 2048 * 4096, out);
  // 5) out = expm(mixed) in place (Jacobi + WMMA reconstruction)
  jacobi_fn_kernel<<<2048, 256, 0, stream>>>(out, nullptr, 1);
}